// BSplineField1d_13821204759217
// MI455X (gfx1250) — compile-verified
//
#include <hip/hip_runtime.h>

// 1D cubic B-spline field evaluation for MI455X (gfx1250).
// Memory-bound: 8 B/point HBM traffic (~268 MB total), ~11.5us floor at 23.3 TB/s.
// phi table (32 KB) is DMA'd into LDS once per workgroup via the CDNA5
// Tensor Data Mover (tensor_load_to_lds + s_wait_tensorcnt), then all
// per-point gathers are ds_load_b32 from LDS. The shared-array pointer is
// passed INTO the asm so the TDM write is visible to alias analysis and the
// LDS reads are not dead-coded.

#ifndef USE_TDM
#define USE_TDM 1
#endif

static constexpr int MAX_CP_LDS = 8192;

__device__ __forceinline__ int clampi(int v, int hi) {
    v = v < 0 ? 0 : v;
    return v > hi ? hi : v;
}

// Single-point cubic B-spline evaluation. TblF is a lambda indexing either
// the __shared__ table (ds_load) or the global table (global_load).
template <class TblF>
__device__ __forceinline__ float bspline1(float tval, TblF&& tbl,
                                          float dx, float inv_dx, float origin, int ncm1) {
    const float k6 = 1.0f / 6.0f;
    float tt  = (tval - origin) - dx;
    float q   = tt * inv_dx;          // reference divides; mul-by-recip (C2-continuous, benign)
    float f   = floorf(q);
    int   idx = (int)f;
    float u   = q - f;

    float u2  = u * u;
    float u3  = u2 * u;
    float omu = 1.0f - u;
    float w0  = omu * omu * omu * k6;
    float w1  = (3.0f * u3 - 6.0f * u2 + 4.0f) * k6;
    float w2  = (-3.0f * u3 + 3.0f * u2 + 3.0f * u + 1.0f) * k6;
    float w3  = u3 * k6;

    int i0 = clampi(idx,     ncm1);
    int i1 = clampi(idx + 1, ncm1);
    int i2 = clampi(idx + 2, ncm1);
    int i3 = clampi(idx + 3, ncm1);

    return w0 * tbl(i0) + w1 * tbl(i1) + w2 * tbl(i2) + w3 * tbl(i3);
}

// Grid-stride float4 streaming loop over the query points.
template <class TblF>
__device__ __forceinline__ void eval_stream(const float* __restrict__ t,
                                            float* __restrict__ out, int n, TblF tbl,
                                            float dx, float inv_dx, float origin, int ncm1) {
    const long gstride = (long)gridDim.x * (long)blockDim.x;
    const long tid0    = (long)blockIdx.x * (long)blockDim.x + threadIdx.x;
    const long n4      = (long)(n >> 2);

    const float4* __restrict__ t4 = (const float4*)t;
    float4* __restrict__ o4       = (float4*)out;

#pragma unroll 2
    for (long i = tid0; i < n4; i += gstride) {
        // Speculative prefetch of next tile of the t stream -> global_prefetch_b8.
        __builtin_prefetch(&t4[i + gstride], 0, 0);
        float4 tv = t4[i];
        float4 r;
        r.x = bspline1(tv.x, tbl, dx, inv_dx, origin, ncm1);
        r.y = bspline1(tv.y, tbl, dx, inv_dx, origin, ncm1);
        r.z = bspline1(tv.z, tbl, dx, inv_dx, origin, ncm1);
        r.w = bspline1(tv.w, tbl, dx, inv_dx, origin, ncm1);
        o4[i] = r;
    }
    // Scalar tail (n not divisible by 4).
    for (long i = (n4 << 2) + tid0; i < (long)n; i += gstride) {
        out[i] = bspline1(t[i], tbl, dx, inv_dx, origin, ncm1);
    }
}

__global__ __launch_bounds__(256) void bspline_field1d_kernel(const float* __restrict__ t,
                                                              const float* __restrict__ phi,
                                                              float* __restrict__ out,
                                                              int n, int num_cp) {
    __shared__ float s_phi[MAX_CP_LDS];
    const bool use_lds = (num_cp <= MAX_CP_LDS);

    if (use_lds) {
#if USE_TDM
        // One wave issues a TDM load of the whole table into LDS.
        // D# per CDNA5 ISA (2-group form, <=2D tensor).
        if (threadIdx.x == 0) {
            typedef unsigned uint4v __attribute__((ext_vector_type(4)));
            typedef unsigned uint8v __attribute__((ext_vector_type(8)));
            const unsigned long long ga = (unsigned long long)phi;
            const unsigned lo  = (unsigned)ga;
            const unsigned hi  = (unsigned)(ga >> 32);
            const unsigned ncp = (unsigned)num_cp;
            // group0: count=1 | lds_addr=0 (s_phi is at LDS base) | global_addr[56:0] | type=2
            uint4v g0 = { 1u,
                          0u,
                          lo,
                          (hi & 0x01FFFFFFu) | (2u << 30) };
            // group1: wg_mask=0, data_size=2 (4B); tensor_dim0=ncp; tensor_dim1=1;
            //         tile_dim0=ncp; tile_dim1=1; tensor_dim0_stride=ncp
            uint8v g1 = { (2u << 16),
                          (ncp & 0xFFFFu) << 16,            // tensor_dim0[15:0] @ bits[63:48]
                          (ncp >> 16) | (1u << 16),         // tensor_dim0[31:16], tensor_dim1[15:0]=1
                          (ncp & 0xFFFFu) << 16,            // tensor_dim1[31:16]=0, tile_dim0=ncp
                          1u,                               // tile_dim1=1, tile_dim2=0
                          ncp,                              // tensor_dim0_stride[31:0]
                          0u,                               // stride0[47:32]=0, tensor_dim1_stride lo=0
                          0u };
            // %2 (the LDS destination pointer) is unused in the template but makes
            // s_phi reachable from the asm, so the "memory" clobber is understood
            // to write the shared array (prevents DCE of the ds_load gathers).
            asm volatile("tensor_load_to_lds %0, %1"
                         :
                         : "s"(g0), "s"(g1), "v"(&s_phi[0])
                         : "memory");
            __builtin_amdgcn_s_wait_tensorcnt(0);
        }
#else
        // Cooperative vectorized copy fallback: global_load_b128 + ds_store_b128.
        const float4* __restrict__ p4 = (const float4*)phi;
        float4* s4 = (float4*)s_phi;
        for (int i = threadIdx.x; i < (num_cp >> 2); i += blockDim.x) s4[i] = p4[i];
        for (int i = (num_cp & ~3) + threadIdx.x; i < num_cp; i += blockDim.x)
            s_phi[i] = phi[i];
#endif
    }
    __syncthreads();

    const float dx     = 2.0f / (float)(num_cp - 3);
    const float origin = -1.0f - dx;
    const float inv_dx = 1.0f / dx;
    const int   ncm1   = num_cp - 1;

    if (use_lds) {
        // Direct shared-array indexing keeps addrspace(3) -> ds_load_b32 gathers.
        eval_stream(t, out, n,
                    [&](int i) -> float { return s_phi[i]; },
                    dx, inv_dx, origin, ncm1);
    } else {
        // L2-resident fallback (table > LDS budget).
        eval_stream(t, out, n,
                    [&](int i) -> float { return phi[i]; },
                    dx, inv_dx, origin, ncm1);
    }
}

extern "C" void kernel_launch(void* const* d_in, const int* in_sizes, int n_in,
                              void* d_out, int out_size, void* d_ws, size_t ws_size,
                              hipStream_t stream) {
    const float* t   = (const float*)d_in[0];
    const float* phi = (const float*)d_in[1];
    float* out       = (float*)d_out;
    const int n      = in_sizes[0];
    const int num_cp = in_sizes[1];

    const int threads = 256;                 // 8 wave32 waves per block
    int n4 = n >> 2;
    int blocks = (n4 + threads - 1) / threads;
    if (blocks > 4096) blocks = 4096;        // amortize LDS table fill; grid-stride covers rest
    if (blocks < 1) blocks = 1;

    bspline_field1d_kernel<<<blocks, threads, 0, stream>>>(t, phi, out, n, num_cp);
}